// DMLNegHead_73349451481901
// MI455X (gfx1250) — compile-verified
//
#include <hip/hip_runtime.h>
#include <math.h>

// ---------------------------------------------------------------------------
// DMLNegHead for MI455X (gfx1250).  fp32 WMMA (V_WMMA_F32_16X16X4_F32) GEMM of
// [131072 pixels x 128ch] x [128 x 384 padded prototype cols], fused with all
// elementwise / reduction epilogue.  All global stores coalesced via LDS
// transpose tiles.  Memory-bound: ~364 MB HBM traffic -> ~16us at 23.3 TB/s.
// ---------------------------------------------------------------------------

typedef float v2f __attribute__((ext_vector_type(2)));
typedef float v8f __attribute__((ext_vector_type(8)));

#define O_    81
#define CCH   128
#define HWSZ  16384          // 128*128

// padded prototype table: rows 0..95 = pos (81 valid), rows 96+k*96+o = neg
#define PROT_ROWS   384
#define PROT_STRIDE 132      // LDS row pad: 132 % 64 = 4 -> conflict-free
#define PROT_FLOATS (PROT_ROWS * PROT_STRIDE)   // 50688
#define MINW_FLOATS (8 * 16 * 96)               // 12288
#define TB_FLOATS   (8 * 16 * 17)               // 2176
#define PB_FLOATS   (8 * 16 * 96)               // 12288
#define SMEM_FLOATS (PROT_FLOATS + MINW_FLOATS + TB_FLOATS + PB_FLOATS)
#define SMEM_BYTES  (SMEM_FLOATS * 4)           // 309760 B < 320KB WGP LDS

// ======================= prep kernels (tiny) ================================

// grid 96 x 128: row o -> normalized (rep_fc_w + rep_fc_b), zeros for o>=81
__global__ void dml_prep_reps(const float* __restrict__ w,
                              const float* __restrict__ b,
                              float* __restrict__ protos) {
  const int o = blockIdx.x, c = threadIdx.x;
  __shared__ float red[128];
  float v = 0.f;
  if (o < O_) v = w[o * 128 + c] + b[o * 128 + c];
  red[c] = v * v;
  __syncthreads();
  for (int s = 64; s > 0; s >>= 1) {
    if (c < s) red[c] += red[c + s];
    __syncthreads();
  }
  const float invn = 1.0f / fmaxf(sqrtf(red[0]), 1e-12f);
  protos[o * 128 + c] = v * invn;
}

// grid 96 x 128: out[o][j] = relu?( sum_c in[o][c]*W[j][c] + bias[j] )
__global__ void dml_prep_mm(const float* __restrict__ in,
                            const float* __restrict__ W,
                            const float* __restrict__ bias,
                            float* __restrict__ out, int relu) {
  const int o = blockIdx.x, j = threadIdx.x;
  __shared__ float hrow[128];
  hrow[j] = in[o * 128 + j];
  __syncthreads();
  const float* wr = W + (size_t)j * 128;
  float acc = bias[j];
#pragma unroll 8
  for (int c = 0; c < 128; ++c) acc += hrow[c] * wr[c];
  if (relu) acc = fmaxf(acc, 0.f);
  out[o * 128 + j] = acc;
}

// grid 96 x 128: normalize row, mask pad rows to zero, store into protos table
__global__ void dml_prep_norm(const float* __restrict__ in,
                              float* __restrict__ protos, int dstRowBase) {
  const int o = blockIdx.x, c = threadIdx.x;
  __shared__ float red[128];
  float v = (o < O_) ? in[o * 128 + c] : 0.f;
  red[c] = v * v;
  __syncthreads();
  for (int s = 64; s > 0; s >>= 1) {
    if (c < s) red[c] += red[c + s];
    __syncthreads();
  }
  const float invn = 1.0f / fmaxf(sqrtf(red[0]), 1e-12f);
  protos[(size_t)(dstRowBase + o) * 128 + c] = v * invn;
}

// ============================ main kernel ===================================
// 1024 blocks x 256 threads (8 waves).  Wave w handles 16 consecutive pixels.
__global__ void dml_main(const float* __restrict__ x,
                         const float* __restrict__ protos_g,
                         float* __restrict__ out_cls,
                         float* __restrict__ out_csn,
                         float* __restrict__ out_dist,
                         float* __restrict__ out_dneg,
                         float* __restrict__ out_pori) {
  extern __shared__ float smem[];
  float* sProt = smem;                    // [384][132]
  float* sMin  = smem + PROT_FLOATS;      // per-wave [16][96]
  float* sTb   = sMin + MINW_FLOATS;      // per-wave [16][17] transpose tile
  float* sPb   = sTb + TB_FLOATS;         // per-wave [16][96] probs staging

  const int tid = threadIdx.x;

  // cooperative prototype load (row-major 128 -> padded stride 132)
  for (int idx = tid; idx < PROT_ROWS * 128; idx += 256)
    sProt[(idx >> 7) * PROT_STRIDE + (idx & 127)] = protos_g[idx];
  for (int idx = tid; idx < MINW_FLOATS; idx += 256) sMin[idx] = 3.0e38f;
  __syncthreads();

  const int wave = tid >> 5;
  const int lane = tid & 31;
  const int m16  = lane & 15;   // pixel-in-tile (A rows) / N column (B,C,D)
  const int half = lane >> 4;   // K-pair selector per ISA f32 layouts
  const int kb   = half << 1;

  float* minw = sMin + wave * (16 * 96);
  float* tb   = sTb + wave * (16 * 17);
  float* pb   = sPb + wave * (16 * 96);

  const int P0 = blockIdx.x * 128 + wave * 16;  // first pixel of wave tile
  const int n  = P0 >> 14;                      // / 16384
  const int hw = P0 & (HWSZ - 1);

  // ---- load A fragments (16 pixels x 128 ch) + per-pixel L2 norm ----------
  // lane holds pixel m16, channels with (c % 4) in {kb, kb+1}
  float areg[64];
  float ss = 0.f;
  const float* xp = x + ((size_t)n * CCH * HWSZ + (size_t)(hw + m16));
#pragma unroll
  for (int kk = 0; kk < 32; ++kk) {
    const int c0 = 4 * kk + kb;
    const float v0 = xp[(size_t)c0 * HWSZ];
    const float v1 = xp[(size_t)(c0 + 1) * HWSZ];
    areg[2 * kk]     = v0;
    areg[2 * kk + 1] = v1;
    ss += v0 * v0 + v1 * v1;
  }
  ss += __shfl_xor(ss, 16, 32);  // partner lane holds complementary channels
  const float invn = 1.0f / fmaxf(sqrtf(ss), 1e-12f);
#pragma unroll
  for (int i = 0; i < 64; ++i) areg[i] *= invn;

  const float INV = 2.0f;  // 1/(2*SIGMA^2), SIGMA=0.5

  // ---- 24 N-tiles: neg tiles (6..23) first, then pos tiles (0..5) ---------
  for (int tt = 0; tt < 24; ++tt) {
    const int t = (tt < 18) ? (tt + 6) : (tt - 18);

    v8f acc = {0.f, 0.f, 0.f, 0.f, 0.f, 0.f, 0.f, 0.f};
    const float* bbase = sProt + (size_t)(t * 16 + m16) * PROT_STRIDE + kb;
#pragma unroll
    for (int kk = 0; kk < 32; ++kk) {
      v2f a;
      a[0] = areg[2 * kk];
      a[1] = areg[2 * kk + 1];
      const float2 bv = *reinterpret_cast<const float2*>(bbase + 4 * kk);
      v2f b;
      b[0] = bv.x;
      b[1] = bv.y;
      acc = __builtin_amdgcn_wmma_f32_16x16x4_f32(
          false, a, false, b, (short)0, acc, false, false);
    }

    // stage d = sqrt(relu(2 - 2*dot)) into transpose tile
    // C/D layout: VGPR i, lane -> M = i + 8*half, N = m16
#pragma unroll
    for (int i = 0; i < 8; ++i) {
      const float dot = acc[i];
      const float d = sqrtf(fmaxf(2.f - 2.f * dot, 0.f));
      tb[(i + 8 * half) * 17 + m16] = d;
    }
    __syncthreads();

    if (t >= 6) {  // ---- negative-prototype tile ----
      const int j0 = (t - 6) * 16;   // 0..287, aligned inside one k-block
      const int kN = j0 / 96;
      const int ob = j0 % 96;
#pragma unroll
      for (int cc = 0; cc < 8; ++cc) {
        const int col = 2 * cc + half;            // lane-pix = m16 now
        const float dn = tb[m16 * 17 + col];
        const int o = ob + col;                   // unique (pix,o) per lane
        minw[m16 * 96 + o] = fminf(minw[m16 * 96 + o], dn);
        if (o < O_) {
          out_dneg[(((size_t)(n * O_ + o)) * 3 + kN) * HWSZ +
                   (size_t)(hw + m16)] = dn;       // coalesced over pixels
        }
      }
    } else {  // ---- positive-prototype tile ----
      const int ob = t * 16;
#pragma unroll
      for (int cc = 0; cc < 8; ++cc) {
        const int col = 2 * cc + half;
        const int o = ob + col;
        const float d = tb[m16 * 17 + col];
        float pr = 0.f;
        if (o < O_) {
          const size_t oi = ((size_t)(n * O_ + o)) * HWSZ + (size_t)(hw + m16);
          out_dist[oi] = d;
          out_pori[oi] = __expf(-d * d * INV);
          const float mn = minw[m16 * 96 + o];
          const float dd = d + 0.3f * fmaxf(2.0f - mn, 0.f);  // BETA, NEG_SCOPE
          pr = __expf(-dd * dd * INV);
        }
        pb[m16 * 96 + o] = pr;                    // pads stay exactly 0
      }
    }
    __syncthreads();

    if (tt == 17) {  // all neg tiles done: cls_score_neg = exp(-min_k(d)^2*inv)
#pragma unroll 4
      for (int cc = 0; cc < 48; ++cc) {
        const int col = 2 * cc + half;
        if (col < O_) {
          const float mn = minw[m16 * 96 + col];
          out_csn[((size_t)(n * O_ + col)) * HWSZ + (size_t)(hw + m16)] =
              __expf(-mn * mn * INV);
        }
      }
    }
  }

  // ---- cls_score: normalize probs over the 81 classes per pixel -----------
  float part = 0.f;
#pragma unroll 8
  for (int q = 0; q < 48; ++q) part += pb[m16 * 96 + half * 48 + q];
  part += __shfl_xor(part, 16, 32);
  const float invt = 1.0f / part;
#pragma unroll 4
  for (int cc = 0; cc < 48; ++cc) {
    const int col = 2 * cc + half;
    if (col < O_) {
      out_cls[((size_t)(n * O_ + col)) * HWSZ + (size_t)(hw + m16)] =
          pb[m16 * 96 + col] * invt;
    }
  }
}

// ============================= launcher =====================================

extern "C" void kernel_launch(void* const* d_in, const int* in_sizes, int n_in,
                              void* d_out, int out_size, void* d_ws,
                              size_t ws_size, hipStream_t stream) {
  const float* x     = (const float*)d_in[0];
  const float* rep_w = (const float*)d_in[1];
  const float* rep_b = (const float*)d_in[2];
  const float* neg_w = (const float*)d_in[3];
  const float* neg_b = (const float*)d_in[4];

  float* ws     = (float*)d_ws;
  float* protos = ws;                         // [384][128]
  float* hA     = ws + 384 * 128;             // [96][128]
  float* hB     = hA + 96 * 128;              // [96][128]

  float* out = (float*)d_out;
  const size_t PER = (size_t)8 * O_ * HWSZ;   // 10,616,832
  float* out_cls  = out;
  float* out_csn  = out + PER;
  float* out_dist = out + 2 * PER;
  float* out_dneg = out + 3 * PER;            // 3*PER elements
  float* out_pori = out + 6 * PER;

  // prototype prep: normalize reps, run 3x 3-layer MLP, normalize neg reps
  dml_prep_reps<<<96, 128, 0, stream>>>(rep_w, rep_b, protos);
  for (int k = 0; k < 3; ++k) {
    dml_prep_mm<<<96, 128, 0, stream>>>(
        protos, neg_w + (size_t)(k * 3 + 0) * 16384, neg_b + (k * 3 + 0) * 128,
        hA, 1);
    dml_prep_mm<<<96, 128, 0, stream>>>(
        hA, neg_w + (size_t)(k * 3 + 1) * 16384, neg_b + (k * 3 + 1) * 128,
        hB, 1);
    dml_prep_mm<<<96, 128, 0, stream>>>(
        hB, neg_w + (size_t)(k * 3 + 2) * 16384, neg_b + (k * 3 + 2) * 128,
        hA, 0);
    dml_prep_norm<<<96, 128, 0, stream>>>(hA, protos, 96 + k * 96);
  }

  // main fused WMMA kernel: 131072 pixels / 128 per block
  (void)hipFuncSetAttribute(reinterpret_cast<const void*>(dml_main),
                            hipFuncAttributeMaxDynamicSharedMemorySize,
                            SMEM_BYTES);
  dml_main<<<1024, 256, SMEM_BYTES, stream>>>(x, protos, out_cls, out_csn,
                                              out_dist, out_dneg, out_pori);
}